// TreeDecoder_17935783428632
// MI455X (gfx1250) — compile-verified
//
#include <hip/hip_runtime.h>
#include <math.h>

// ---------------------------------------------------------------------------
// TreeDecoder for MI455X (gfx1250), fp32 end-to-end via V_WMMA_F32_16X16X4_F32
// ---------------------------------------------------------------------------

typedef float v2f __attribute__((ext_vector_type(2)));
typedef float v8f __attribute__((ext_vector_type(8)));

#define BDIM   64
#define CDIM   64
#define HDIM   256
#define LDIM   64
#define PDIM   32
#define NNODES 2048
#define NREAL  2047
#define K1     192      // 3*C
#define K2     768      // 3*H
#define KM     320      // H + L
#define NTILES (NNODES / 16)   // 128
#define CNT    ((float)(HDIM * NNODES))

__device__ __forceinline__ v8f wmma_f32(v2f a, v2f b, v8f c) {
  // D = A(16x4 f32) x B(4x16 f32) + C(16x16 f32)
  return __builtin_amdgcn_wmma_f32_16x16x4_f32(
      /*neg_a=*/false, a, /*neg_b=*/false, b,
      /*c_mod=*/(short)0, c, /*reuse_a=*/false, /*reuse_b=*/false);
}

// ---------------------------------------------------------------------------
// Repack [H x Kdim] row-major weights into WMMA A-fragment-major order:
// pack[((htile*nkk + kk)*32 + lane)*2 + {0,1}] = W[htile*16 + (lane&15)][kk*4 + (lane>>4)*2 + {0,1}]
// so a wave's A-load per K-step is one contiguous 256B burst.
// ---------------------------------------------------------------------------
__global__ void pack_w_kernel(const float* __restrict__ W,
                              float* __restrict__ out, int Kdim) {
  int nkk = Kdim >> 2;
  int total = (HDIM / 16) * nkk * 32;
  int tid = blockIdx.x * blockDim.x + threadIdx.x;
  if (tid >= total) return;
  int lane = tid & 31;
  int kk   = (tid >> 5) % nkk;
  int ht   = tid / (nkk * 32);
  int m  = lane & 15;
  int kb = (lane >> 4) << 1;
  int h  = ht * 16 + m;
  int k0 = kk * 4 + kb;
  out[(size_t)tid * 2 + 0] = W[(size_t)h * Kdim + k0];
  out[(size_t)tid * 2 + 1] = W[(size_t)h * Kdim + k0 + 1];
}

// ---------------------------------------------------------------------------
// Deterministic per-batch reduction of 128 per-tile (sum, sumsq) partials.
// part layout: [B*128] sums, then [B*128] sumsqs.  stats: [B] sums, [B] ssq.
// ---------------------------------------------------------------------------
__global__ void reduce_stats_kernel(const float* __restrict__ part,
                                    float* __restrict__ stats) {
  __shared__ float rs[128];
  __shared__ float rq[128];
  int b = blockIdx.x, t = threadIdx.x;
  rs[t] = part[(size_t)b * NTILES + t];
  rq[t] = part[(size_t)BDIM * NTILES + (size_t)b * NTILES + t];
  __syncthreads();
  for (int s = 64; s > 0; s >>= 1) {
    if (t < s) { rs[t] += rs[t + s]; rq[t] += rq[t + s]; }
    __syncthreads();
  }
  if (t == 0) { stats[b] = rs[0]; stats[BDIM + b] = rq[0]; }
}

// ---------------------------------------------------------------------------
// Tree conv: out[b,h,n] = sum_{c,j} in[b,c,children[b,3(n-1)+j]] * W[h,3c+j] + bias[h]
// (n==0 column forced to zero).  Optionally fuses TreeLayerNorm+ReLU of the
// INPUT (statsIn != null).  Emits per-block (sum, sumsq) partials of OUTPUT.
// Block: 256 threads = 8 waves; block owns 16 output nodes x all 256 h.
// ---------------------------------------------------------------------------
__global__ __launch_bounds__(256)
void tree_conv_kernel(const float* __restrict__ xin,
                      const int* __restrict__ children,
                      const float* __restrict__ packW,
                      const float* __restrict__ bias,
                      float* __restrict__ xout,
                      const float* __restrict__ statsIn,  // [B] sum, [B] ssq (or null)
                      float* __restrict__ partOut,        // [B*128] sum, [B*128] ssq
                      int Kdim) {
  __shared__ float G[K2 * 16];     // gathered B-matrix, [Kdim][16], worst case 48KB
  __shared__ int   chi[48];        // children triples for the 16 nodes
  __shared__ float redS[256];
  __shared__ float redQ[256];

  const int tile = blockIdx.x;     // 0..127 -> nodes [tile*16, tile*16+16)
  const int b    = blockIdx.y;
  const int tid  = threadIdx.x;
  const int lane = tid & 31;
  const int wv   = tid >> 5;
  const int Cin  = Kdim / 3;
  const bool normIn = (statsIn != nullptr);

  float mean = 0.f, rstd = 1.f;
  if (normIn) {
    float s = statsIn[b], q = statsIn[BDIM + b];
    mean = s / CNT;
    float var = (q - s * s / CNT) / (CNT - 1.0f);
    var = var > 0.f ? var : 0.f;
    rstd = 1.0f / (sqrtf(var) + 1e-5f);
  }

  if (tid < 48) {
    int nl = tid / 3, j = tid - nl * 3;
    int n = tile * 16 + nl;
    int c = 0;
    if (n >= 1) c = children[(size_t)b * 3 * NREAL + 3 * (n - 1) + j];
    chi[tid] = c;
  }
  __syncthreads();

  // Cooperative gather of G[k][nloc] (with fused norm+relu of the input layer).
  const float* xb = xin + (size_t)b * Cin * NNODES;
  for (int e = tid; e < Kdim * 16; e += 256) {
    int k = e >> 4, nl = e & 15;
    int n = tile * 16 + nl;
    float v = 0.f;
    if (n >= 1) {
      int c = k / 3, j = k - c * 3;
      v = xb[(size_t)c * NNODES + chi[nl * 3 + j]];
      if (normIn) { v = (v - mean) * rstd; v = v > 0.f ? v : 0.f; }
    }
    G[k * 16 + nl] = v;
  }
  __syncthreads();

  // Each wave owns 2 h-tiles (8 waves x 2 x 16 = 256 rows).
  const int nkk  = Kdim >> 2;
  const int nloc = lane & 15;
  const int kb   = (lane >> 4) << 1;
  const float* pw0 = packW + (size_t)(wv * 2 + 0) * nkk * 64;
  const float* pw1 = packW + (size_t)(wv * 2 + 1) * nkk * 64;

  v8f acc0 = {}; v8f acc1 = {};
  for (int kk = 0; kk < nkk; ++kk) {
    int k0 = kk * 4 + kb;
    v2f bf; bf.x = G[k0 * 16 + nloc]; bf.y = G[(k0 + 1) * 16 + nloc];
    v2f a0 = *(const v2f*)(pw0 + (size_t)kk * 64 + lane * 2);
    v2f a1 = *(const v2f*)(pw1 + (size_t)kk * 64 + lane * 2);
    acc0 = wmma_f32(a0, bf, acc0);
    acc1 = wmma_f32(a1, bf, acc1);
  }

  // Epilogue: bias, n==0 zero column, store, local sum/sumsq.
  const int mB = (lane < 16) ? 0 : 8;
  const int n  = tile * 16 + nloc;
  float lsum = 0.f, lsq = 0.f;
  float* xo = xout + (size_t)b * HDIM * NNODES;
#pragma unroll
  for (int r = 0; r < 8; ++r) {
    int h0 = (wv * 2 + 0) * 16 + mB + r;
    int h1 = (wv * 2 + 1) * 16 + mB + r;
    float v0 = acc0[r] + bias[h0];
    float v1 = acc1[r] + bias[h1];
    if (n == 0) { v0 = 0.f; v1 = 0.f; }
    xo[(size_t)h0 * NNODES + n] = v0;
    xo[(size_t)h1 * NNODES + n] = v1;
    lsum += v0 + v1;
    lsq  += v0 * v0 + v1 * v1;
  }

  redS[tid] = lsum; redQ[tid] = lsq;
  __syncthreads();
  for (int s = 128; s > 0; s >>= 1) {
    if (tid < s) { redS[tid] += redS[tid + s]; redQ[tid] += redQ[tid + s]; }
    __syncthreads();
  }
  if (tid == 0) {
    partOut[(size_t)b * NTILES + tile] = redS[0];
    partOut[(size_t)BDIM * NTILES + (size_t)b * NTILES + tile] = redQ[0];
  }
}

// ---------------------------------------------------------------------------
// Fused MLP per 16-node tile:
//   xz[n,k] = relu(norm(x2[b,k,n]))  (k<256)  |  z[b,k-256]  (k>=256)
//   h  = relu(xz @ W1 + b1)   -> LDS [16][257]
//   out= h @ W2 + b2          -> [B, N, 32]
// ---------------------------------------------------------------------------
__global__ __launch_bounds__(256)
void mlp_kernel(const float* __restrict__ x2raw,
                const float* __restrict__ z,
                const float* __restrict__ w1, const float* __restrict__ b1,
                const float* __restrict__ w2, const float* __restrict__ b2,
                const float* __restrict__ stats2,   // [B] sum, [B] ssq
                float* __restrict__ out) {
  __shared__ float hmat[16 * 257];   // padded stride avoids LDS bank conflicts

  const int tile = blockIdx.x, b = blockIdx.y;
  const int tid = threadIdx.x, lane = tid & 31, wv = tid >> 5;

  float s = stats2[b], q = stats2[BDIM + b];
  float mean = s / CNT;
  float var  = (q - s * s / CNT) / (CNT - 1.0f);
  var = var > 0.f ? var : 0.f;
  float rstd = 1.0f / (sqrtf(var) + 1e-5f);

  const int nloc = lane & 15;
  const int kb   = (lane >> 4) << 1;
  const float* xb = x2raw + (size_t)b * HDIM * NNODES + (size_t)tile * 16;
  const float* zb = z + (size_t)b * LDIM;

  // ---- phase 1: h = relu(xz @ W1 + b1), each wave -> 2 h-tiles of 16 ----
  const int h0a = (wv * 2 + 0) * 16;
  const int h1a = (wv * 2 + 1) * 16;
  v8f acc0 = {}; v8f acc1 = {};
  for (int kk = 0; kk < KM / 4; ++kk) {
    int k0 = kk * 4 + kb;
    v2f a;
    if (k0 < HDIM) {
      float v = xb[(size_t)k0 * NNODES + nloc];
      v = (v - mean) * rstd; a.x = v > 0.f ? v : 0.f;
    } else a.x = zb[k0 - HDIM];
    if (k0 + 1 < HDIM) {
      float v = xb[(size_t)(k0 + 1) * NNODES + nloc];
      v = (v - mean) * rstd; a.y = v > 0.f ? v : 0.f;
    } else a.y = zb[k0 + 1 - HDIM];

    v2f bf0, bf1;
    bf0.x = w1[(size_t)k0 * HDIM + h0a + nloc];
    bf0.y = w1[(size_t)(k0 + 1) * HDIM + h0a + nloc];
    bf1.x = w1[(size_t)k0 * HDIM + h1a + nloc];
    bf1.y = w1[(size_t)(k0 + 1) * HDIM + h1a + nloc];
    acc0 = wmma_f32(a, bf0, acc0);
    acc1 = wmma_f32(a, bf1, acc1);
  }
  const int mB = (lane < 16) ? 0 : 8;
#pragma unroll
  for (int r = 0; r < 8; ++r) {
    int nl = mB + r;
    float v0 = acc0[r] + b1[h0a + nloc];
    float v1 = acc1[r] + b1[h1a + nloc];
    hmat[nl * 257 + h0a + nloc] = v0 > 0.f ? v0 : 0.f;
    hmat[nl * 257 + h1a + nloc] = v1 > 0.f ? v1 : 0.f;
  }
  __syncthreads();

  // ---- phase 2: out = h @ W2 + b2 ; two 16x16 output tiles (waves 0,1) ----
  if (wv < 2) {
    const int p0 = wv * 16;
    v8f acc = {};
    for (int kk = 0; kk < HDIM / 4; ++kk) {
      int k0 = kk * 4 + kb;
      v2f a;  a.x = hmat[nloc * 257 + k0];
              a.y = hmat[nloc * 257 + k0 + 1];
      v2f bf; bf.x = w2[(size_t)k0 * PDIM + p0 + nloc];
              bf.y = w2[(size_t)(k0 + 1) * PDIM + p0 + nloc];
      acc = wmma_f32(a, bf, acc);
    }
    float* ob = out + ((size_t)b * NNODES + (size_t)tile * 16) * PDIM;
#pragma unroll
    for (int r = 0; r < 8; ++r) {
      int nl = mB + r;
      ob[(size_t)nl * PDIM + p0 + nloc] = acc[r] + b2[p0 + nloc];
    }
  }
}

// ---------------------------------------------------------------------------
extern "C" void kernel_launch(void* const* d_in, const int* in_sizes, int n_in,
                              void* d_out, int out_size, void* d_ws, size_t ws_size,
                              hipStream_t stream) {
  const float* node_feats = (const float*)d_in[0];
  const float* z          = (const float*)d_in[1];
  const int*   children   = (const int*)d_in[2];
  const float* conv1_w    = (const float*)d_in[3];
  const float* conv1_b    = (const float*)d_in[4];
  const float* conv2_w    = (const float*)d_in[5];
  const float* conv2_b    = (const float*)d_in[6];
  const float* mlp_w1     = (const float*)d_in[7];
  const float* mlp_b1     = (const float*)d_in[8];
  const float* mlp_w2     = (const float*)d_in[9];
  const float* mlp_b2     = (const float*)d_in[10];
  float* out = (float*)d_out;
  float* ws  = (float*)d_ws;

  // Workspace layout (floats)
  float* X1  = ws;                                   // [B,H,N] raw conv1 out (128 MB)
  float* X2  = X1 + (size_t)BDIM * HDIM * NNODES;    // [B,H,N] raw conv2 out (128 MB)
  float* PW1 = X2 + (size_t)BDIM * HDIM * NNODES;    // packed conv1 weights
  float* PW2 = PW1 + (size_t)HDIM * K1;              // packed conv2 weights
  float* P1  = PW2 + (size_t)HDIM * K2;              // conv1 (sum,ssq) partials
  float* P2  = P1 + (size_t)2 * BDIM * NTILES;       // conv2 (sum,ssq) partials
  float* ST1 = P2 + (size_t)2 * BDIM * NTILES;       // conv1 batch stats [2*B]
  float* ST2 = ST1 + 2 * BDIM;                       // conv2 batch stats [2*B]

  pack_w_kernel<<<(HDIM * K1 / 2 + 255) / 256, 256, 0, stream>>>(conv1_w, PW1, K1);
  pack_w_kernel<<<(HDIM * K2 / 2 + 255) / 256, 256, 0, stream>>>(conv2_w, PW2, K2);

  dim3 grid(NTILES, BDIM);
  tree_conv_kernel<<<grid, 256, 0, stream>>>(node_feats, children, PW1, conv1_b,
                                             X1, nullptr, P1, K1);
  reduce_stats_kernel<<<BDIM, 128, 0, stream>>>(P1, ST1);
  tree_conv_kernel<<<grid, 256, 0, stream>>>(X1, children, PW2, conv2_b,
                                             X2, ST1, P2, K2);
  reduce_stats_kernel<<<BDIM, 128, 0, stream>>>(P2, ST2);
  mlp_kernel<<<grid, 256, 0, stream>>>(X2, z, mlp_w1, mlp_b1, mlp_w2, mlp_b2,
                                       ST2, out);
}